// TradeFlowRGCN_65352222376641
// MI455X (gfx1250) — compile-verified
//
#include <hip/hip_runtime.h>
#include <hip/hip_bf16.h>

#define HIDD 64
#define NREL 3
#define Q1C 5000.0f
#define Q2C 10000.0f

typedef __attribute__((ext_vector_type(2))) float v2f;
typedef __attribute__((ext_vector_type(8))) float v8f;

// ---------------------------------------------------------------------------
// 1) classify edges by distance threshold
// ---------------------------------------------------------------------------
__global__ void etype_kernel(const float* __restrict__ ea, int* __restrict__ et,
                             long long nEdges) {
    long long e = (long long)blockIdx.x * blockDim.x + threadIdx.x;
    if (e >= nEdges) return;
    float d = ea[e * 6];
    et[e] = (d > Q1C ? 1 : 0) + (d > Q2C ? 1 : 0);
}

// ---------------------------------------------------------------------------
// 2) layer-1 scatter: sums over x[src] (din=3) per (rel,dst) + edge counts
// ---------------------------------------------------------------------------
__global__ void scatter_l1(const float* __restrict__ x,
                           const long long* __restrict__ ei,
                           const int* __restrict__ et,
                           float* __restrict__ sums,   // [3][N][64] (first 3 used)
                           float* __restrict__ counts, // [3][N]
                           int nNodes, long long nEdges) {
    long long e = (long long)blockIdx.x * blockDim.x + threadIdx.x;
    if (e >= nEdges) return;
    int r = et[e];
    long long s = ei[e];
    long long d = ei[nEdges + e];
    float* base = sums + ((size_t)r * nNodes + (size_t)d) * HIDD;
    const float* xs = x + (size_t)s * 3;
    atomicAdd(base + 0, xs[0]);
    atomicAdd(base + 1, xs[1]);
    atomicAdd(base + 2, xs[2]);
    atomicAdd(counts + (size_t)r * nNodes + (size_t)d, 1.0f);
}

// ---------------------------------------------------------------------------
// 3) layer-1 node update (din=3): scalar, one thread per (node, out-col)
// ---------------------------------------------------------------------------
__global__ void node_l1(const float* __restrict__ x,
                        const float* __restrict__ sums,
                        const float* __restrict__ counts,
                        const float* __restrict__ root0, // [3][64]
                        const float* __restrict__ w0,    // [3][3][64]
                        const float* __restrict__ b0,    // [64]
                        float* __restrict__ h0,          // [N][64]
                        int nNodes) {
    long long idx = (long long)blockIdx.x * blockDim.x + threadIdx.x;
    if (idx >= (long long)nNodes * HIDD) return;
    int n = (int)(idx >> 6);
    int j = (int)(idx & 63);
    float acc = b0[j];
    float xv0 = x[(size_t)n * 3 + 0];
    float xv1 = x[(size_t)n * 3 + 1];
    float xv2 = x[(size_t)n * 3 + 2];
    acc += xv0 * root0[0 * HIDD + j] + xv1 * root0[1 * HIDD + j] + xv2 * root0[2 * HIDD + j];
#pragma unroll
    for (int r = 0; r < NREL; ++r) {
        float c = counts[(size_t)r * nNodes + n];
        float inv = 1.0f / fmaxf(c, 1.0f);
        const float* sb = sums + ((size_t)r * nNodes + n) * HIDD;
        const float* wb = w0 + (size_t)r * 3 * HIDD;
        acc += (sb[0] * inv) * wb[0 * HIDD + j];
        acc += (sb[1] * inv) * wb[1 * HIDD + j];
        acc += (sb[2] * inv) * wb[2 * HIDD + j];
    }
    h0[(size_t)n * HIDD + j] = fmaxf(acc, 0.0f);
}

// ---------------------------------------------------------------------------
// 4) 64-wide scatter: one thread per (edge, feature); coalesced f32 atomics
// ---------------------------------------------------------------------------
__global__ void scatter64(const float* __restrict__ h,
                          const long long* __restrict__ ei,
                          const int* __restrict__ et,
                          float* __restrict__ sums, // [3][N][64]
                          int nNodes, long long nEdges) {
    long long idx = (long long)blockIdx.x * blockDim.x + threadIdx.x;
    if (idx >= nEdges * HIDD) return;
    long long e = idx >> 6;
    int k = (int)(idx & 63);
    int r = et[e];
    long long s = ei[e];
    long long d = ei[nEdges + e];
    float v = h[(size_t)s * HIDD + k];
    atomicAdd(sums + ((size_t)r * nNodes + (size_t)d) * HIDD + k, v);
}

// ---------------------------------------------------------------------------
// 5) layers 2/3 node update via WMMA f32 16x16x4.
//    WG = 128 threads = 4 waves; 16 nodes per WG; wave w owns out cols
//    [16w, 16w+16). A-sources: h_in tile + 3 normalized relation means,
//    staged in LDS (stride 66 to dodge bank conflicts).
// ---------------------------------------------------------------------------
#define ASTRIDE 66
__global__ __launch_bounds__(128)
void node_rgcn64_wmma(const float* __restrict__ h_in,   // [N][64]
                      const float* __restrict__ sums,   // [3][N][64]
                      const float* __restrict__ counts, // [3][N]
                      const float* __restrict__ root,   // [64][64]
                      const float* __restrict__ W,      // [3][64][64]
                      const float* __restrict__ bias,   // [64]
                      float* __restrict__ h_out,        // [N][64]
                      int nNodes) {
    __shared__ float ldsA[4][16][ASTRIDE];
    const int tid = threadIdx.x;
    const int nb = blockIdx.x * 16;

    // stage A operands: h tile and the 3 relation means (sum / max(count,1))
    for (int i = tid; i < 16 * HIDD; i += 128) {
        int m = i >> 6;
        int k = i & 63;
        int node = nb + m;
        float hv = 0.0f, m0 = 0.0f, m1 = 0.0f, m2 = 0.0f;
        if (node < nNodes) {
            hv = h_in[(size_t)node * HIDD + k];
            float c0 = fmaxf(counts[(size_t)0 * nNodes + node], 1.0f);
            float c1 = fmaxf(counts[(size_t)1 * nNodes + node], 1.0f);
            float c2 = fmaxf(counts[(size_t)2 * nNodes + node], 1.0f);
            m0 = sums[((size_t)0 * nNodes + node) * HIDD + k] / c0;
            m1 = sums[((size_t)1 * nNodes + node) * HIDD + k] / c1;
            m2 = sums[((size_t)2 * nNodes + node) * HIDD + k] / c2;
        }
        ldsA[0][m][k] = hv;
        ldsA[1][m][k] = m0;
        ldsA[2][m][k] = m1;
        ldsA[3][m][k] = m2;
    }
    __syncthreads();

    const int lane = tid & 31;
    const int wave = tid >> 5;
    const int ncol = wave * 16;
    const int ln = lane & 15;          // M index for A frag / N index for B frag
    const int ko = (lane >> 4) << 1;   // K sub-offset: lanes 0-15 -> {0,1}, 16-31 -> {2,3}

    v8f acc = {};
#pragma unroll
    for (int s = 0; s < 4; ++s) {
        const float* Bmat = (s == 0) ? root : (W + (size_t)(s - 1) * HIDD * HIDD);
        const float* arow = &ldsA[s][ln][0];
#pragma unroll
        for (int kb = 0; kb < HIDD; kb += 4) {
            v2f a, b;
            a.x = arow[kb + ko];
            a.y = arow[kb + ko + 1];
            b.x = Bmat[(size_t)(kb + ko) * HIDD + ncol + ln];
            b.y = Bmat[(size_t)(kb + ko + 1) * HIDD + ncol + ln];
            acc = __builtin_amdgcn_wmma_f32_16x16x4_f32(
                false, a, false, b, (short)0, acc, false, false);
        }
    }

    // epilogue: bias + ReLU, scatter D fragment (VGPR v -> row v or v+8)
    float bj = bias[ncol + ln];
#pragma unroll
    for (int v = 0; v < 8; ++v) {
        int m = (lane < 16) ? v : (v + 8);
        int node = nb + m;
        if (node < nNodes) {
            float d = fmaxf(acc[v] + bj, 0.0f);
            h_out[(size_t)node * HIDD + ncol + ln] = d;
        }
    }
}

// ---------------------------------------------------------------------------
// 6) decoder MLP: 134 -> 32 -> 16 -> 1 per edge; weights staged in LDS
// ---------------------------------------------------------------------------
__global__ __launch_bounds__(256)
void decoder_kernel(const float* __restrict__ h,
                    const long long* __restrict__ ei,
                    const float* __restrict__ ea,
                    const float* __restrict__ dw1, const float* __restrict__ db1,
                    const float* __restrict__ dw2, const float* __restrict__ db2,
                    const float* __restrict__ dw3, const float* __restrict__ db3,
                    float* __restrict__ out, long long nEdges) {
    __shared__ float s_dw1[134 * 32];
    __shared__ float s_dw2[32 * 16];
    __shared__ float s_dw3[16];
    __shared__ float s_db1[32];
    __shared__ float s_db2[16];
    for (int i = threadIdx.x; i < 134 * 32; i += 256) s_dw1[i] = dw1[i];
    for (int i = threadIdx.x; i < 32 * 16; i += 256) s_dw2[i] = dw2[i];
    if (threadIdx.x < 16) s_dw3[threadIdx.x] = dw3[threadIdx.x];
    if (threadIdx.x < 32) s_db1[threadIdx.x] = db1[threadIdx.x];
    if (threadIdx.x < 16) s_db2[threadIdx.x] = db2[threadIdx.x];
    __syncthreads();

    long long e = (long long)blockIdx.x * blockDim.x + threadIdx.x;
    if (e >= nEdges) return;
    long long s = ei[e];
    long long d = ei[nEdges + e];
    const float* hs = h + (size_t)s * HIDD;
    const float* hd = h + (size_t)d * HIDD;

    float z1[32];
#pragma unroll
    for (int j = 0; j < 32; ++j) z1[j] = s_db1[j];
    for (int k = 0; k < 64; ++k) {
        float a = hs[k];
        const float* w = &s_dw1[k * 32];
#pragma unroll
        for (int j = 0; j < 32; ++j) z1[j] += a * w[j];
    }
    for (int k = 0; k < 64; ++k) {
        float a = hd[k];
        const float* w = &s_dw1[(64 + k) * 32];
#pragma unroll
        for (int j = 0; j < 32; ++j) z1[j] += a * w[j];
    }
    for (int k = 0; k < 6; ++k) {
        float a = ea[e * 6 + k];
        const float* w = &s_dw1[(128 + k) * 32];
#pragma unroll
        for (int j = 0; j < 32; ++j) z1[j] += a * w[j];
    }
    float z2[16];
#pragma unroll
    for (int j = 0; j < 16; ++j) z2[j] = s_db2[j];
    for (int k = 0; k < 32; ++k) {
        float a = fmaxf(z1[k], 0.0f);
        const float* w = &s_dw2[k * 16];
#pragma unroll
        for (int j = 0; j < 16; ++j) z2[j] += a * w[j];
    }
    float o = db3[0];
#pragma unroll
    for (int k = 0; k < 16; ++k) o += fmaxf(z2[k], 0.0f) * s_dw3[k];
    out[e] = o;
}

// ---------------------------------------------------------------------------
// host-side orchestration
// ---------------------------------------------------------------------------
extern "C" void kernel_launch(void* const* d_in, const int* in_sizes, int n_in,
                              void* d_out, int out_size, void* d_ws, size_t ws_size,
                              hipStream_t stream) {
    const float*     x     = (const float*)d_in[0];
    const long long* ei    = (const long long*)d_in[1];
    const float*     ea    = (const float*)d_in[2];
    const float*     w0    = (const float*)d_in[3];
    const float*     root0 = (const float*)d_in[4];
    const float*     b0    = (const float*)d_in[5];
    const float*     w1    = (const float*)d_in[6];
    const float*     root1 = (const float*)d_in[7];
    const float*     b1    = (const float*)d_in[8];
    const float*     w2    = (const float*)d_in[9];
    const float*     root2 = (const float*)d_in[10];
    const float*     b2    = (const float*)d_in[11];
    const float*     dw1   = (const float*)d_in[12];
    const float*     db1   = (const float*)d_in[13];
    const float*     dw2   = (const float*)d_in[14];
    const float*     db2   = (const float*)d_in[15];
    const float*     dw3   = (const float*)d_in[16];
    const float*     db3   = (const float*)d_in[17];

    const int nNodes = in_sizes[0] / 3;
    const long long nEdges = (long long)in_sizes[1] / 2;

    // ws sub-allocation (256B aligned)
    char* ws = (char*)d_ws;
    size_t off = 0;
    auto take = [&](size_t bytes) -> char* {
        char* p = ws + off;
        off = (off + bytes + 255) & ~(size_t)255;
        return p;
    };
    int*   etype  = (int*)  take((size_t)nEdges * sizeof(int));
    float* counts = (float*)take((size_t)NREL * nNodes * sizeof(float));
    float* sums   = (float*)take((size_t)NREL * nNodes * HIDD * sizeof(float));
    float* hA     = (float*)take((size_t)nNodes * HIDD * sizeof(float));
    float* hB     = (float*)take((size_t)nNodes * HIDD * sizeof(float));

    const size_t sumsBytes = (size_t)NREL * nNodes * HIDD * sizeof(float);

    // edge types + counts (counts are layer-invariant)
    hipMemsetAsync(counts, 0, (size_t)NREL * nNodes * sizeof(float), stream);
    hipMemsetAsync(sums, 0, sumsBytes, stream);
    {
        long long g = (nEdges + 255) / 256;
        etype_kernel<<<(dim3)(unsigned)g, 256, 0, stream>>>(ea, etype, nEdges);
        scatter_l1<<<(dim3)(unsigned)g, 256, 0, stream>>>(x, ei, etype, sums, counts,
                                                          nNodes, nEdges);
    }
    // layer 1 node update -> hA
    {
        long long g = ((long long)nNodes * HIDD + 255) / 256;
        node_l1<<<(dim3)(unsigned)g, 256, 0, stream>>>(x, sums, counts, root0, w0, b0,
                                                       hA, nNodes);
    }
    // layer 2: scatter hA, WMMA node update -> hB
    hipMemsetAsync(sums, 0, sumsBytes, stream);
    {
        long long g = (nEdges * HIDD + 255) / 256;
        scatter64<<<(dim3)(unsigned)g, 256, 0, stream>>>(hA, ei, etype, sums, nNodes, nEdges);
        node_rgcn64_wmma<<<(nNodes + 15) / 16, 128, 0, stream>>>(hA, sums, counts, root1,
                                                                 w1, b1, hB, nNodes);
    }
    // layer 3: scatter hB, WMMA node update -> hA (reuse)
    hipMemsetAsync(sums, 0, sumsBytes, stream);
    {
        long long g = (nEdges * HIDD + 255) / 256;
        scatter64<<<(dim3)(unsigned)g, 256, 0, stream>>>(hB, ei, etype, sums, nNodes, nEdges);
        node_rgcn64_wmma<<<(nNodes + 15) / 16, 128, 0, stream>>>(hB, sums, counts, root2,
                                                                 w2, b2, hA, nNodes);
    }
    // decoder
    {
        long long g = (nEdges + 255) / 256;
        decoder_kernel<<<(dim3)(unsigned)g, 256, 0, stream>>>(hA, ei, ea, dw1, db1, dw2,
                                                              db2, dw3, db3,
                                                              (float*)d_out, nEdges);
    }
}